// VGAE_47579647705163
// MI455X (gfx1250) — compile-verified
//
#include <hip/hip_runtime.h>
#include <hip/hip_bf16.h>

// VGAE fused forward for MI455X (gfx1250, wave32, WMMA).
// B=32, N=1024, F=256, D=128.
// Strategy: fp32 WMMA (V_WMMA_F32_16X16X4_F32) everywhere -- problem is
// HBM-bound on the 128MB adjacency read (~10us at 23.3TB/s), so full fp32
// matches reference numerics at no perf cost. Logits [B,N,N] (128MB) are
// never materialized: BCE is reduced inside the z@z^T tiles, and since
// logits are symmetric, only upper-triangular tiles are computed (2x less
// WMMA work -> GEMM2 compute safely under the HBM roof).

typedef __attribute__((ext_vector_type(2))) float v2f;
typedef __attribute__((ext_vector_type(8))) float v8f;

#define BQ 32
#define NQ 1024
#define FQ 256
#define DQ 128

__device__ __forceinline__ float waveReduceSum(float v) {
  // wave32 butterfly reduction
  #pragma unroll
  for (int off = 16; off >= 1; off >>= 1)
    v += __shfl_xor(v, off, 32);
  return v;
}

__device__ __forceinline__ v8f wmma_f32_k4(v2f a, v2f b, v8f c) {
  return __builtin_amdgcn_wmma_f32_16x16x4_f32(
      /*neg_a=*/false, a, /*neg_b=*/false, b,
      /*c_mod=*/(short)0, c, /*reuse_a=*/false, /*reuse_b=*/false);
}

// ---------------------------------------------------------------------------
// Kernel 1: mean/log_std heads + reparameterization + KL partials.
// One wave computes one 16x16 tile of (mean, log_std) over K=F=256.
// grid = (D/16, B*N/16), block = 32 (one wave -> EXEC all-ones for WMMA).
// ---------------------------------------------------------------------------
__global__ __launch_bounds__(32) void vgae_encode(
    const float* __restrict__ X,   // [B*N, F]
    const float* __restrict__ Wm,  // [F, D]
    const float* __restrict__ bm,  // [D]
    const float* __restrict__ Ws,  // [F, D]
    const float* __restrict__ bs,  // [D]
    const float* __restrict__ eps, // [B*N, D]
    float* __restrict__ zout,      // [B*N, D]
    float* __restrict__ klPartial) // [gridDim.y * gridDim.x]
{
  const int lane = threadIdx.x;
  const int half = lane >> 4;   // 0: K pair {0,1} / rows 0-7; 1: {2,3} / rows 8-15
  const int idx  = lane & 15;   // M (for A) or N (for B/C/D)
  const int row0 = blockIdx.y << 4;
  const int col0 = blockIdx.x << 4;

  const float* Xr = X + (size_t)(row0 + idx) * FQ + 2 * half;
  v8f accM = {};
  v8f accS = {};

  #pragma unroll 4
  for (int k0 = 0; k0 < FQ; k0 += 4) {
    v2f a = *(const v2f*)(Xr + k0);                      // A: (M=idx, K=k0+2h..+1)
    const float* wmp = Wm + (size_t)(k0 + 2 * half) * DQ + col0 + idx;
    const float* wsp = Ws + (size_t)(k0 + 2 * half) * DQ + col0 + idx;
    v2f bmv; bmv.x = wmp[0]; bmv.y = wmp[DQ];            // B: (K, N=idx)
    v2f bsv; bsv.x = wsp[0]; bsv.y = wsp[DQ];
    accM = wmma_f32_k4(a, bmv, accM);
    accS = wmma_f32_k4(a, bsv, accS);
  }

  const float bmc = bm[col0 + idx];
  const float bsc = bs[col0 + idx];
  const int   col = col0 + idx;
  float klsum = 0.f;

  #pragma unroll
  for (int r = 0; r < 8; ++r) {
    const int row = row0 + (half << 3) + r;              // C/D row mapping
    float m  = accM[r] + bmc;
    float ls = accS[r] + bsc;
    float e  = eps[(size_t)row * DQ + col];
    float z  = fmaf(expf(ls), e, m);
    zout[(size_t)row * DQ + col] = z;
    // 1 + 2*log_std - mean^2 - exp(2*log_std)
    klsum += 1.f + 2.f * ls - m * m - expf(2.f * ls);
  }

  klsum = waveReduceSum(klsum);
  if (lane == 0)
    klPartial[blockIdx.y * gridDim.x + blockIdx.x] = klsum;
}

// ---------------------------------------------------------------------------
// Kernel 2: fused z@z^T + BCE reduction, exploiting logits symmetry.
// One wave per 16x16 logits tile, K=D=128 (32 WMMA steps), but only
// upper-triangular tiles (ti <= tj) are computed. Off-diagonal tiles fold in
// the mirrored block's BCE term using L(m,n) = L(n,m) and lsig(-x)=lsig(x)-x:
//   term(n,m) + term(m,n) = 2*t - (2 - a(n,m) - a(m,n)) * L.
// Lower-triangular blocks write a 0.0 partial and exit (block-uniform branch,
// EXEC stays all-ones for WMMA). z stays hot in 192MB L2; adj is single-use
// and read non-temporally, exactly one pass over 128MB.
// grid = (N/16, N/16, B), block = 32.
// ---------------------------------------------------------------------------
__global__ __launch_bounds__(32) void vgae_bce(
    const float* __restrict__ zbuf, // [B, N, D]
    const float* __restrict__ adj,  // [B, N, N]
    float* __restrict__ bcePartial) // [B * (N/16)^2]
{
  const int lane = threadIdx.x;
  const int half = lane >> 4;
  const int idx  = lane & 15;
  const int b  = blockIdx.z;
  const int ti = blockIdx.y;
  const int tj = blockIdx.x;
  const size_t slot = ((size_t)b * gridDim.y + ti) * gridDim.x + tj;

  if (ti > tj) {               // mirrored tile handled by (tj, ti)
    if (lane == 0) bcePartial[slot] = 0.f;
    return;
  }

  const int n0 = ti << 4;
  const int m0 = tj << 4;
  const float* zb = zbuf + (size_t)b * NQ * DQ;
  const float* za = zb + (size_t)(n0 + idx) * DQ + 2 * half; // rows of A
  const float* zc = zb + (size_t)(m0 + idx) * DQ + 2 * half; // rows forming B=z^T

  v8f acc = {};
  #pragma unroll 8
  for (int k0 = 0; k0 < DQ; k0 += 4) {
    v2f a  = *(const v2f*)(za + k0);
    v2f bv = *(const v2f*)(zc + k0);
    acc = wmma_f32_k4(a, bv, acc);
  }

  const float* adjb = adj + (size_t)b * NQ * NQ;
  const int    col  = m0 + idx;
  float s = 0.f;

  if (ti == tj) {
    // Diagonal tile: single contribution per element.
    #pragma unroll
    for (int r = 0; r < 8; ++r) {
      const int row = n0 + (half << 3) + r;
      float L = acc[r];
      float a = __builtin_nontemporal_load(adjb + (size_t)row * NQ + col);
      // t = log_sigmoid(L) (stable); a*lsig(L) + (1-a)*lsig(-L) = t - (1-a)*L
      float t = fminf(L, 0.f) - log1pf(expf(-fabsf(L)));
      s += t - (1.f - a) * L;
    }
  } else {
    // Off-diagonal: fold (n,m) and the mirrored (m,n) element together.
    #pragma unroll
    for (int r = 0; r < 8; ++r) {
      const int row = n0 + (half << 3) + r;
      float L  = acc[r];
      float a1 = __builtin_nontemporal_load(adjb + (size_t)row * NQ + col);
      float a2 = __builtin_nontemporal_load(adjb + (size_t)col * NQ + row);
      float t  = fminf(L, 0.f) - log1pf(expf(-fabsf(L)));
      s += 2.f * t - (2.f - a1 - a2) * L;
    }
  }

  s = waveReduceSum(s);
  if (lane == 0) bcePartial[slot] = s;
}

// ---------------------------------------------------------------------------
// Kernel 3: deterministic final reduction. 32 waves, wave w owns graph w.
// bce partials: 4096/graph (lower-tri slots are 0); kl partials: 512/graph.
// ---------------------------------------------------------------------------
__global__ __launch_bounds__(1024) void vgae_finalize(
    const float* __restrict__ bcePartial, // [32*4096]
    const float* __restrict__ klPartial,  // [32*512]
    float* __restrict__ outLoss)
{
  __shared__ float per_graph[BQ];
  const int lane = threadIdx.x & 31;
  const int w    = threadIdx.x >> 5; // graph id

  float s = 0.f;
  const float* bp = bcePartial + (size_t)w * 4096;
  for (int i = lane; i < 4096; i += 32) s += bp[i];

  float k = 0.f;
  const float* kp = klPartial + (size_t)w * 512;
  for (int i = lane; i < 512; i += 32) k += kp[i];

  s = waveReduceSum(s);
  k = waveReduceSum(k);

  if (lane == 0) {
    float bce_b = -s / ((float)NQ * (float)NQ);
    float kl_b  = -0.5f * k / ((float)NQ * (float)DQ);
    per_graph[w] = bce_b + kl_b;
  }
  __syncthreads();
  if (threadIdx.x == 0) {
    float t = 0.f;
    #pragma unroll
    for (int g = 0; g < BQ; ++g) t += per_graph[g];
    outLoss[0] = t / (float)BQ;
  }
}

// ---------------------------------------------------------------------------
extern "C" void kernel_launch(void* const* d_in, const int* in_sizes, int n_in,
                              void* d_out, int out_size, void* d_ws, size_t ws_size,
                              hipStream_t stream) {
  (void)in_sizes; (void)n_in; (void)out_size; (void)ws_size;

  const float* X   = (const float*)d_in[0]; // node_feats [B,N,F]
  const float* adj = (const float*)d_in[1]; // [B,N,N]
  const float* eps = (const float*)d_in[2]; // [B,N,D]
  const float* Wm  = (const float*)d_in[3]; // [F,D]
  const float* bm  = (const float*)d_in[4]; // [D]
  const float* Ws  = (const float*)d_in[5]; // [F,D]
  const float* bs  = (const float*)d_in[6]; // [D]

  float* out     = (float*)d_out;
  const size_t nX = (size_t)BQ * NQ * FQ;   // 8,388,608
  const size_t nZ = (size_t)BQ * NQ * DQ;   // 4,194,304
  float* outX    = out;
  float* outZ    = out + nX;
  float* outLoss = out + nX + nZ;

  float* klPartial  = (float*)d_ws;         // 2048*8 = 16384 floats
  float* bcePartial = klPartial + 16384;    // 32*64*64 = 131072 floats (~576KB total)

  // Output 1: node_feats passthrough (d2d copy, graph-capture safe).
  hipMemcpyAsync(outX, X, nX * sizeof(float), hipMemcpyDeviceToDevice, stream);

  // Output 2: z (+ KL partials).
  vgae_encode<<<dim3(DQ / 16, (BQ * NQ) / 16, 1), 32, 0, stream>>>(
      X, Wm, bm, Ws, bs, eps, outZ, klPartial);

  // Fused symmetric z@z^T + BCE partials (reads outZ; same stream -> ordered).
  vgae_bce<<<dim3(NQ / 16, NQ / 16, BQ), 32, 0, stream>>>(outZ, adj, bcePartial);

  // Output 3: scalar loss.
  vgae_finalize<<<1, 1024, 0, stream>>>(bcePartial, klPartial, outLoss);
}